// BiDirAttention_28149215658289
// MI455X (gfx1250) — compile-verified
//
#include <hip/hip_runtime.h>
#include <hip/hip_bf16.h>

// Problem constants (match reference setup_inputs)
constexpr int Bc = 32;
constexpr int Nc = 2048;
constexpr int Mc = 512;
constexpr int Dc = 128;
constexpr float NEGC = -1.0e9f;

typedef float v2f __attribute__((ext_vector_type(2)));
typedef float v8f __attribute__((ext_vector_type(8)));

__device__ __forceinline__ float waveMax(float v) {
    #pragma unroll
    for (int o = 16; o; o >>= 1) v = fmaxf(v, __shfl_xor(v, o, 32));
    return v;
}
__device__ __forceinline__ float waveSum(float v) {
    #pragma unroll
    for (int o = 16; o; o >>= 1) v += __shfl_xor(v, o, 32);
    return v;
}

// ---------------------------------------------------------------------------
// Phase 1: one block per (batch, 16-row tile of N).
//   S = seq1_tile @ seq2^T  (f32 WMMA 16x16x4)
//   masked softmax over m -> a2 (global NT store + LDS), s1 = row max -> ws
//   out1 = a2 @ seq2        (f32 WMMA 16x16x4), NT store
// ---------------------------------------------------------------------------
__global__ __launch_bounds__(256) void bidir_attn_phase1(
    const float* __restrict__ seq1, const float* __restrict__ seq2,
    const int* __restrict__ m2, float* __restrict__ out_o1,
    float* __restrict__ out_a2, float* __restrict__ ws_s1)
{
    __shared__ float As[16 * 130];   // seq1 tile, padded stride
    __shared__ float Sl[16 * 516];   // logits -> a2, padded stride
    __shared__ float m2f[Mc];

    const int nt = blockIdx.x;
    const int b  = blockIdx.y;
    const int tid = threadIdx.x;
    const int lane = tid & 31;
    const int w = tid >> 5;          // wave id 0..7
    const int n0 = nt * 16;

    const float* s1b = seq1 + ((size_t)b * Nc + n0) * Dc;
    const float* s2b = seq2 + (size_t)b * Mc * Dc;

    // stage seq1 tile + m2
    for (int i = tid; i < 16 * Dc; i += 256)
        As[(i >> 7) * 130 + (i & 127)] = s1b[i];
    for (int i = tid; i < Mc; i += 256)
        m2f[i] = (float)m2[(size_t)b * Mc + i];
    __syncthreads();

    const int col  = lane & 15;          // N-column within 16x16 tile (and A row)
    const int krow = (lane >> 4) * 2;    // K sub-offset per lane half
    const int rbase = (lane >> 4) * 8;   // C/D row base per lane half

    // ---- GEMM1: each wave computes 4 column tiles of S ----
    const float* Arow = As + col * 130 + krow;
    #pragma unroll
    for (int t = 0; t < 4; ++t) {
        const int mt  = w + t * 8;       // 0..31
        const int m0c = mt * 16;
        const float* Brow = s2b + (size_t)(m0c + col) * Dc + krow;
        v8f acc = {};
        #pragma unroll 8
        for (int kk = 0; kk < Dc; kk += 4) {
            v2f a = *(const v2f*)(Arow + kk);
            v2f bb = *(const v2f*)(Brow + kk);
            acc = __builtin_amdgcn_wmma_f32_16x16x4_f32(
                false, a, false, bb, (short)0, acc, false, false);
        }
        const float negadd = (1.0f - m2f[m0c + col]) * NEGC;  // masked logit addend
        #pragma unroll
        for (int v = 0; v < 8; ++v)
            Sl[(rbase + v) * 516 + m0c + col] = acc[v] + negadd;
    }
    __syncthreads();

    // ---- masked softmax over m; wave w handles rows w and w+8 ----
    float* a2g = out_a2 + ((size_t)b * Nc + n0) * Mc;
    #pragma unroll
    for (int rr = 0; rr < 2; ++rr) {
        const int r = w + rr * 8;
        float vals[16];
        float mx = -3.4e38f;
        #pragma unroll
        for (int i = 0; i < 16; ++i) {
            vals[i] = Sl[r * 516 + lane + i * 32];
            mx = fmaxf(mx, vals[i]);
        }
        mx = waveMax(mx);
        float sum = 0.f;
        #pragma unroll
        for (int i = 0; i < 16; ++i) { vals[i] = __expf(vals[i] - mx); sum += vals[i]; }
        sum = waveSum(sum);
        const float inv = 1.0f / sum;
        #pragma unroll
        for (int i = 0; i < 16; ++i) {
            const float a = vals[i] * inv;
            Sl[r * 516 + lane + i * 32] = a;
            // write-only 134MB stream: non-temporal so seq1/seq2 stay L2-resident
            __builtin_nontemporal_store(a, a2g + (size_t)r * Mc + lane + i * 32);
        }
        // rowmax of masked logits == s1 (masked entries <= NEG in both formulas)
        if (lane == 0) ws_s1[(size_t)b * Nc + n0 + r] = mx;
    }
    __syncthreads();

    // ---- GEMM2: out1_tile = a2_tile[16x512] @ seq2[512x128]; wave w -> d-tile w ----
    {
        const int d0 = w * 16;
        const float* Arow2 = Sl + col * 516 + krow;
        const float* Bbase = s2b + (size_t)krow * Dc + d0 + col;
        v8f acc = {};
        #pragma unroll 8
        for (int kk = 0; kk < Mc; kk += 4) {
            v2f a = *(const v2f*)(Arow2 + kk);
            v2f bb = { Bbase[(size_t)kk * Dc], Bbase[(size_t)kk * Dc + Dc] };
            acc = __builtin_amdgcn_wmma_f32_16x16x4_f32(
                false, a, false, bb, (short)0, acc, false, false);
        }
        float* o1 = out_o1 + ((size_t)b * Nc + n0) * Dc;
        #pragma unroll
        for (int v = 0; v < 8; ++v)
            __builtin_nontemporal_store(acc[v],
                o1 + (size_t)(rbase + v) * Dc + d0 + col);
    }
}

// ---------------------------------------------------------------------------
// Phase 2: one block per batch. a1 = softmax_n(s1 + (1-m1)*NEG);
//          seq2_seq1[b,d] = sum_n a1[n] * seq1[b,n,d]
// ---------------------------------------------------------------------------
__global__ __launch_bounds__(256) void bidir_attn_phase2(
    const float* __restrict__ seq1, const int* __restrict__ m1,
    const float* __restrict__ ws_s1, float* __restrict__ out_a1,
    float* __restrict__ out_o3)
{
    __shared__ float a1s[Nc];
    __shared__ float red[8];
    __shared__ float part[Dc];

    const int b = blockIdx.x;
    const int tid = threadIdx.x;
    const int lane = tid & 31;
    const int w = tid >> 5;
    const size_t base = (size_t)b * Nc;

    float lmax = -3.4e38f;
    for (int n = tid; n < Nc; n += 256) {
        float x = ws_s1[base + n] + (1.0f - (float)m1[base + n]) * NEGC;
        a1s[n] = x;
        lmax = fmaxf(lmax, x);
    }
    lmax = waveMax(lmax);
    if (lane == 0) red[w] = lmax;
    __syncthreads();
    float mx = red[0];
    #pragma unroll
    for (int i = 1; i < 8; ++i) mx = fmaxf(mx, red[i]);

    float lsum = 0.f;
    for (int n = tid; n < Nc; n += 256) {
        float e = __expf(a1s[n] - mx);
        a1s[n] = e;
        lsum += e;
    }
    lsum = waveSum(lsum);
    __syncthreads();               // all reads of red (max) done
    if (lane == 0) red[w] = lsum;
    __syncthreads();
    float sum = 0.f;
    #pragma unroll
    for (int i = 0; i < 8; ++i) sum += red[i];
    const float inv = 1.0f / sum;

    for (int n = tid; n < Nc; n += 256) {
        float a = a1s[n] * inv;
        a1s[n] = a;
        out_a1[base + n] = a;
    }
    __syncthreads();

    // weighted sum over n: 256 threads = 128 d-cols x 2 n-halves
    const int d = tid & 127;
    const int h = tid >> 7;
    const float* sp = seq1 + (base + (size_t)h * 1024) * Dc + d;
    float acc = 0.f;
    for (int n = 0; n < 1024; ++n)
        acc += a1s[h * 1024 + n] * sp[(size_t)n * Dc];
    if (h == 1) part[d] = acc;
    __syncthreads();
    if (h == 0) out_o3[(size_t)b * Dc + d] = acc + part[d];
}

extern "C" void kernel_launch(void* const* d_in, const int* in_sizes, int n_in,
                              void* d_out, int out_size, void* d_ws, size_t ws_size,
                              hipStream_t stream) {
    (void)in_sizes; (void)n_in; (void)out_size; (void)ws_size;
    const float* seq1 = (const float*)d_in[0];
    const float* seq2 = (const float*)d_in[1];
    const int*   m1   = (const int*)d_in[2];
    const int*   m2   = (const int*)d_in[3];

    float* out = (float*)d_out;
    float* out_o1 = out;                                   // [B,N,D]
    float* out_a2 = out_o1 + (size_t)Bc * Nc * Dc;         // [B,N,M]
    float* out_o3 = out_a2 + (size_t)Bc * Nc * Mc;         // [B,D]
    float* out_a1 = out_o3 + (size_t)Bc * Dc;              // [B,N]

    float* ws_s1 = (float*)d_ws;                           // [B,N] row maxima

    dim3 grid1(Nc / 16, Bc);
    bidir_attn_phase1<<<grid1, 256, 0, stream>>>(seq1, seq2, m2, out_o1, out_a2, ws_s1);
    bidir_attn_phase2<<<Bc, 256, 0, stream>>>(seq1, m1, ws_s1, out_a1, out_o3);
}